// SimpleRasterizer_23270132809955
// MI455X (gfx1250) — compile-verified
//
#include <hip/hip_runtime.h>
#include <hip/hip_bf16.h>

#define IMG    256
#define KFA    8
#define NVERT  4096
#define NFACE  5000
#define CHUNK  512
#define NPIX   ((size_t)2 * IMG * IMG * KFA)

#if defined(__gfx1250__) && __has_builtin(__builtin_amdgcn_global_load_async_to_lds_b128) && __has_builtin(__builtin_amdgcn_s_wait_asynccnt)
#define USE_ASYNC_LDS 1
#else
#define USE_ASYNC_LDS 0
#endif

#if USE_ASYNC_LDS
typedef __attribute__((ext_vector_type(4))) int v4i_t;
typedef v4i_t __attribute__((address_space(1)))* gv4i_p;   // global int4*
typedef v4i_t __attribute__((address_space(3)))* lv4i_p;   // LDS int4*
#endif

// _seg_d2 replicated with per-op IEEE rounding (no contraction), matching the
// reference's expression order exactly.
__device__ __forceinline__ float seg_d2(float px, float py,
                                        float ax, float ay,
                                        float bx, float by) {
    float dx = __fsub_rn(bx, ax);
    float dy = __fsub_rn(by, ay);
    float l2 = fmaxf(__fadd_rn(__fmul_rn(dx, dx), __fmul_rn(dy, dy)), 1e-12f);
    float tn = __fadd_rn(__fmul_rn(__fsub_rn(px, ax), dx),
                         __fmul_rn(__fsub_rn(py, ay), dy));
    float t  = __fdiv_rn(tn, l2);
    t = fminf(fmaxf(t, 0.0f), 1.0f);
    float qx = __fadd_rn(ax, __fmul_rn(t, dx));
    float qy = __fadd_rn(ay, __fmul_rn(t, dy));
    float ex = __fsub_rn(px, qx);
    float ey = __fsub_rn(py, qy);
    return __fadd_rn(__fmul_rn(ex, ex), __fmul_rn(ey, ey));
}

__global__ __launch_bounds__(256, 1)
void raster_kernel(const float* __restrict__ verts,
                   const int*   __restrict__ faces,
                   float*       __restrict__ out) {
    // 48KB vertex table + 32KB face-constant chunk = 80KB LDS (CDNA5: 320KB/WGP)
    __shared__ float  ldsV[NVERT * 3];
    __shared__ float4 ldsF[CHUNK * 4];

    const int b   = blockIdx.z;
    const int tid = threadIdx.x;

    const float* vbase = verts + (size_t)b * NVERT * 3;
    const int*   fbase = faces + (size_t)b * NFACE * 3;

    // ---- stage the whole vertex table into LDS (async DMA on gfx1250) ----
    {
        const int n4 = (NVERT * 3) / 4;  // 3072 float4s, 16B aligned
#if USE_ASYNC_LDS
        for (int i = tid; i < n4; i += 256) {
            __builtin_amdgcn_global_load_async_to_lds_b128(
                (gv4i_p)((const float4*)vbase + i),
                (lv4i_p)((float4*)ldsV + i),
                0, 0);
        }
        __builtin_amdgcn_s_wait_asynccnt(0);
#else
        for (int i = tid; i < n4; i += 256)
            ((float4*)ldsV)[i] = ((const float4*)vbase)[i];
#endif
    }

    // lane = pixel; wave32 covers a 16x2 pixel footprint (spatial coherence
    // for the divergent "inside" branch). Block = 16x16 pixel tile.
    const int lx = tid & 15, ly = tid >> 4;
    const int ix = blockIdx.x * 16 + lx;
    const int iy = blockIdx.y * 16 + ly;
    // xs = 1 - 2*(i+0.5)/W  (every step exact in fp32 for W=256)
    const float px = __fsub_rn(1.0f, __fdiv_rn(__fmul_rn(2.0f, __fadd_rn((float)ix, 0.5f)), (float)IMG));
    const float py = __fsub_rn(1.0f, __fdiv_rn(__fmul_rn(2.0f, __fadd_rn((float)iy, 0.5f)), (float)IMG));

    // private top-8 (ascending z, stable: earlier face index wins ties)
    float zs[KFA];
    int   fsI[KFA];
#pragma unroll
    for (int k = 0; k < KFA; ++k) { zs[k] = 1e10f; fsI[k] = -1; }

    for (int c0 = 0; c0 < NFACE; c0 += CHUNK) {
        const int n = (NFACE - c0 < CHUNK) ? (NFACE - c0) : CHUNK;
        __syncthreads();
        // ---- cooperatively precompute per-face constants into LDS ----
        for (int j = tid; j < n; j += 256) {
            const int f  = c0 + j;
            const int i0 = fbase[f * 3 + 0], i1 = fbase[f * 3 + 1], i2 = fbase[f * 3 + 2];
            const float x0 = ldsV[i0 * 3 + 0], y0 = ldsV[i0 * 3 + 1], z0 = ldsV[i0 * 3 + 2];
            const float x1 = ldsV[i1 * 3 + 0], y1 = ldsV[i1 * 3 + 1], z1 = ldsV[i1 * 3 + 2];
            const float x2 = ldsV[i2 * 3 + 0], y2 = ldsV[i2 * 3 + 1], z2 = ldsV[i2 * 3 + 2];
            const float area = __fsub_rn(__fmul_rn(__fsub_rn(x1, x0), __fsub_rn(y2, y0)),
                                         __fmul_rn(__fsub_rn(y1, y0), __fsub_rn(x2, x0)));
            const bool valid = (fabsf(area) > 1e-8f) & (z0 > 1e-6f) & (z1 > 1e-6f) & (z2 > 1e-6f);
            // NaN inv_area makes b=NaN -> (b>=0) false -> never inside (== &valid)
            const float invA = valid ? __fdiv_rn(1.0f, area) : __int_as_float(0x7fc00000);
            ldsF[j * 4 + 0] = make_float4(__fsub_rn(x2, x1), __fsub_rn(y2, y1), x1, y1);
            ldsF[j * 4 + 1] = make_float4(__fsub_rn(x0, x2), __fsub_rn(y0, y2), x2, y2);
            ldsF[j * 4 + 2] = make_float4(__fsub_rn(x1, x0), __fsub_rn(y1, y0), x0, y0);
            ldsF[j * 4 + 3] = make_float4(invA, z0, z1, z2);
        }
        __syncthreads();

        // ---- hot loop: every face vs this lane's pixel (LDS broadcast reads) ----
        for (int j = 0; j < n; ++j) {
            const float4 r0 = ldsF[j * 4 + 0];
            const float4 r1 = ldsF[j * 4 + 1];
            const float4 r2 = ldsF[j * 4 + 2];
            const float4 r3 = ldsF[j * 4 + 3];
            const float w0 = __fsub_rn(__fmul_rn(r0.x, __fsub_rn(py, r0.w)),
                                       __fmul_rn(r0.y, __fsub_rn(px, r0.z)));
            const float w1 = __fsub_rn(__fmul_rn(r1.x, __fsub_rn(py, r1.w)),
                                       __fmul_rn(r1.y, __fsub_rn(px, r1.z)));
            const float w2 = __fsub_rn(__fmul_rn(r2.x, __fsub_rn(py, r2.w)),
                                       __fmul_rn(r2.y, __fsub_rn(px, r2.z)));
            const float b0 = __fmul_rn(w0, r3.x);
            const float b1 = __fmul_rn(w1, r3.x);
            const float b2 = __fmul_rn(w2, r3.x);
            if ((b0 >= 0.0f) & (b1 >= 0.0f) & (b2 >= 0.0f)) {
                // exact reference arithmetic so selection keys bit-match top_k
                const float t0 = __fdiv_rn(b0, r3.y);
                const float t1 = __fdiv_rn(b1, r3.z);
                const float t2 = __fdiv_rn(b2, r3.w);
                float den = __fadd_rn(__fadd_rn(t0, t1), t2);
                den = (fabsf(den) > 1e-10f) ? den : 1.0f;
                const float p0 = __fdiv_rn(t0, den);
                const float p1 = __fdiv_rn(t1, den);
                const float p2 = __fdiv_rn(t2, den);
                const float zp = __fadd_rn(__fadd_rn(__fmul_rn(p0, r3.y),
                                                     __fmul_rn(p1, r3.z)),
                                           __fmul_rn(p2, r3.w));
                if (zp < zs[KFA - 1]) {
                    zs[KFA - 1]  = zp;
                    fsI[KFA - 1] = c0 + j;
#pragma unroll
                    for (int k = KFA - 1; k > 0; --k) {
                        const bool sw = zs[k] < zs[k - 1];  // strict: stable on ties
                        const float tz = zs[k]; const int tf = fsI[k];
                        zs[k]      = sw ? zs[k - 1] : zs[k];
                        fsI[k]     = sw ? fsI[k - 1] : fsI[k];
                        zs[k - 1]  = sw ? tz : zs[k - 1];
                        fsI[k - 1] = sw ? tf : fsI[k - 1];
                    }
                }
            }
        }
    }

    // ---- post-pass: full outputs for the <=8 winners per pixel ----
    float* o_p2f = out;
    float* o_z   = out + NPIX;
    float* o_bar = out + 2 * NPIX;
    float* o_d   = out + 5 * NPIX;
    const size_t base = ((((size_t)b * IMG) + (size_t)iy) * IMG + (size_t)ix) * KFA;

#pragma unroll
    for (int k = 0; k < KFA; ++k) {
        float r_face = -1.0f, r_z = -1.0f, r_b0 = -1.0f, r_b1 = -1.0f, r_b2 = -1.0f, r_d = -1.0f;
        const int f = fsI[k];
        if (f >= 0) {
            const int i0 = fbase[f * 3 + 0], i1 = fbase[f * 3 + 1], i2 = fbase[f * 3 + 2];
            const float x0 = ldsV[i0 * 3 + 0], y0 = ldsV[i0 * 3 + 1], z0 = ldsV[i0 * 3 + 2];
            const float x1 = ldsV[i1 * 3 + 0], y1 = ldsV[i1 * 3 + 1], z1 = ldsV[i1 * 3 + 2];
            const float x2 = ldsV[i2 * 3 + 0], y2 = ldsV[i2 * 3 + 1], z2 = ldsV[i2 * 3 + 2];
            const float area = __fsub_rn(__fmul_rn(__fsub_rn(x1, x0), __fsub_rn(y2, y0)),
                                         __fmul_rn(__fsub_rn(y1, y0), __fsub_rn(x2, x0)));
            const float invA = __fdiv_rn(1.0f, area);  // winner => valid
            const float w0 = __fsub_rn(__fmul_rn(__fsub_rn(x2, x1), __fsub_rn(py, y1)),
                                       __fmul_rn(__fsub_rn(y2, y1), __fsub_rn(px, x1)));
            const float w1 = __fsub_rn(__fmul_rn(__fsub_rn(x0, x2), __fsub_rn(py, y2)),
                                       __fmul_rn(__fsub_rn(y0, y2), __fsub_rn(px, x2)));
            const float w2 = __fsub_rn(__fmul_rn(__fsub_rn(x1, x0), __fsub_rn(py, y0)),
                                       __fmul_rn(__fsub_rn(y1, y0), __fsub_rn(px, x0)));
            const float b0 = __fmul_rn(w0, invA);
            const float b1 = __fmul_rn(w1, invA);
            const float b2 = __fmul_rn(w2, invA);
            const float t0 = __fdiv_rn(b0, z0);
            const float t1 = __fdiv_rn(b1, z1);
            const float t2 = __fdiv_rn(b2, z2);
            float den = __fadd_rn(__fadd_rn(t0, t1), t2);
            den = (fabsf(den) > 1e-10f) ? den : 1.0f;
            const float p0 = __fdiv_rn(t0, den);
            const float p1 = __fdiv_rn(t1, den);
            const float p2 = __fdiv_rn(t2, den);
            const float zp = __fadd_rn(__fadd_rn(__fmul_rn(p0, z0), __fmul_rn(p1, z1)),
                                       __fmul_rn(p2, z2));
            const float d2 = fminf(seg_d2(px, py, x0, y0, x1, y1),
                                   fminf(seg_d2(px, py, x1, y1, x2, y2),
                                         seg_d2(px, py, x2, y2, x0, y0)));
            r_face = (float)f; r_z = zp;
            r_b0 = p0; r_b1 = p1; r_b2 = p2;
            r_d = -d2;  // covered => inside => -d2
        }
        o_p2f[base + k] = r_face;
        o_z[base + k]   = r_z;
        o_bar[(base + k) * 3 + 0] = r_b0;
        o_bar[(base + k) * 3 + 1] = r_b1;
        o_bar[(base + k) * 3 + 2] = r_b2;
        o_d[base + k]   = r_d;
    }
}

extern "C" void kernel_launch(void* const* d_in, const int* in_sizes, int n_in,
                              void* d_out, int out_size, void* d_ws, size_t ws_size,
                              hipStream_t stream) {
    (void)in_sizes; (void)n_in; (void)d_ws; (void)ws_size; (void)out_size;
    const float* verts = (const float*)d_in[0];   // (2, 4096, 3) f32
    const int*   faces = (const int*)d_in[1];     // (2, 5000, 3) i32
    float*       out   = (float*)d_out;           // p2f | zbuf | bary | dists
    dim3 grid(IMG / 16, IMG / 16, 2);
    raster_kernel<<<grid, dim3(256, 1, 1), 0, stream>>>(verts, faces, out);
}